// MY_A3TGCN__41901700940307
// MI455X (gfx1250) — compile-verified
//
#include <hip/hip_runtime.h>
#include <math.h>

// A3TGCN forward, MI455X (gfx1250, wave32).
//
// Roofline: edge scatter moves ~0.7GB of mostly-random traffic -> ~30us floor
// @ 23.3 TB/s; memory/atomic bound. Node phase is ~5 GFLOP of fp32 GEMM ->
// V_WMMA_F32_16X16X4_F32 (full fp32, 16x16 tiles, 8 k-steps for K=32).
// H0 stays zero in the reference loop => R-gate is dead code; Hp=(1-Z)*Ht.
//
// v2: column-tile loop hoisted outside the p-loop (halves live B-fragment
// registers -> removes scratch spills seen in v1) and tanh via __expf
// (cuts the libm tanhf VALU expansion).

#define PP 12
#define FF 32

typedef __attribute__((ext_vector_type(2))) float v2f;
typedef __attribute__((ext_vector_type(8))) float v8f;

__global__ __launch_bounds__(256) void k_zero(float* __restrict__ deg, int n) {
  int i = blockIdx.x * 256 + threadIdx.x;
  if (i < n) deg[i] = 0.0f;
}

__global__ __launch_bounds__(256) void k_deg(const int* __restrict__ ei,
                                             const float* __restrict__ ew,
                                             float* __restrict__ deg, int e) {
  int i = blockIdx.x * 256 + threadIdx.x;
  if (i >= e) return;
  atomicAdd(&deg[ei[e + i]], ew[i]);   // dst row of edge_index
}

__global__ __launch_bounds__(256) void k_dinv_init(const float* __restrict__ deg,
                                                   const float* __restrict__ x,
                                                   float* __restrict__ dinv,
                                                   float* __restrict__ agg, int n) {
  int i = blockIdx.x * 256 + threadIdx.x;
  if (i >= n) return;
  float dv = rsqrtf(deg[i] + 1.0f);
  dinv[i] = dv;
  float dv2 = dv * dv;                 // matches reference (dinv*dinv)
  const float4* xv = (const float4*)(x + (size_t)i * PP);
  float4 a = xv[0], b = xv[1], c = xv[2];
  float4* av = (float4*)(agg + (size_t)i * PP);
  av[0] = make_float4(dv2 * a.x, dv2 * a.y, dv2 * a.z, dv2 * a.w);
  av[1] = make_float4(dv2 * b.x, dv2 * b.y, dv2 * b.z, dv2 * b.w);
  av[2] = make_float4(dv2 * c.x, dv2 * c.y, dv2 * c.z, dv2 * c.w);
}

__global__ __launch_bounds__(256) void k_scatter(const int* __restrict__ ei,
                                                 const float* __restrict__ ew,
                                                 const float* __restrict__ dinv,
                                                 const float* __restrict__ x,
                                                 float* __restrict__ agg, int e) {
  int i = blockIdx.x * 256 + threadIdx.x;
  if (i >= e) return;
  const int s = ei[i];
  const int d = ei[e + i];
  const float nrm = dinv[s] * ew[i] * dinv[d];
  const float4* xs = (const float4*)(x + (size_t)s * PP);
  float4 a = xs[0], b = xs[1], c = xs[2];
  float* o = agg + (size_t)d * PP;
  atomicAdd(o + 0,  nrm * a.x); atomicAdd(o + 1,  nrm * a.y);
  atomicAdd(o + 2,  nrm * a.z); atomicAdd(o + 3,  nrm * a.w);
  atomicAdd(o + 4,  nrm * b.x); atomicAdd(o + 5,  nrm * b.y);
  atomicAdd(o + 6,  nrm * b.z); atomicAdd(o + 7,  nrm * b.w);
  atomicAdd(o + 8,  nrm * c.x); atomicAdd(o + 9,  nrm * c.y);
  atomicAdd(o + 10, nrm * c.z); atomicAdd(o + 11, nrm * c.w);
}

// One wave32 per 16-node tile. fp32 WMMA 16x16x4, K=32 in 8 steps, 2 col tiles.
__global__ __launch_bounds__(256) void k_node(
    const float* __restrict__ agg, const float* __restrict__ att_in,
    const float* __restrict__ wc_z, const float* __restrict__ bc_z,
    const float* __restrict__ wc_h, const float* __restrict__ bc_h,
    const float* __restrict__ wl_z, const float* __restrict__ bl_z,
    const float* __restrict__ wl_h, const float* __restrict__ bl_h,
    const float* __restrict__ w_out, const float* __restrict__ b_out,
    float* __restrict__ out, int n, int n_tiles) {
  const int lane  = threadIdx.x & 31;
  const int wave  = threadIdx.x >> 5;
  const int tile  = blockIdx.x * 8 + wave;
  if (tile >= n_tiles) return;          // uniform per wave -> EXEC stays all-ones
  const int m     = lane & 15;          // row / column within 16-wide tile
  const int khalf = lane >> 4;          // which K (A/B) or M (C/D) half

  // softmax(attention), P=12, computed redundantly per lane -> probs in att[]
  float att[PP];
  float amax = -1e30f;
#pragma unroll
  for (int p = 0; p < PP; ++p) { att[p] = att_in[p]; amax = fmaxf(amax, att[p]); }
  float psum = 0.0f;
#pragma unroll
  for (int p = 0; p < PP; ++p) { att[p] = __expf(att[p] - amax); psum += att[p]; }
  const float pinv = 1.0f / psum;
#pragma unroll
  for (int p = 0; p < PP; ++p) att[p] *= pinv;

  // s[p] = agg[node, p] for this lane's row (lanes 16-31 mirror 0-15)
  int node = tile * 16 + m;
  if (node >= n) node = n - 1;          // clamp instead of masking: keep EXEC full
  float s[PP];
  {
    const float4* av = (const float4*)(agg + (size_t)node * PP);
    float4 a0 = av[0], a1 = av[1], a2 = av[2];
    s[0] = a0.x; s[1] = a0.y; s[2]  = a0.z; s[3]  = a0.w;
    s[4] = a1.x; s[5] = a1.y; s[6]  = a1.z; s[7]  = a1.w;
    s[8] = a2.x; s[9] = a2.y; s[10] = a2.z; s[11] = a2.w;
  }

  // A-fragment coefficients: conv[m][k] = s_p * wc[k] + bc[k]
  // A layout (16x4 f32): lanes 0-15 -> K = k0,k0+1 ; lanes 16-31 -> K = k0+2,k0+3
  v2f wzc[8], bzc[8], whc[8], bhc[8];
#pragma unroll
  for (int ks = 0; ks < 8; ++ks) {
    const int kb = 4 * ks + 2 * khalf;
    wzc[ks].x = wc_z[kb]; wzc[ks].y = wc_z[kb + 1];
    bzc[ks].x = bc_z[kb]; bzc[ks].y = bc_z[kb + 1];
    whc[ks].x = wc_h[kb]; whc[ks].y = wc_h[kb + 1];
    bhc[ks].x = bc_h[kb]; bhc[ks].y = bc_h[kb + 1];
  }

  v8f Hacc[2] = { {}, {} };             // C/D layout: VGPR r -> row r (+8*khalf), col m+16t

  // Column-tile loop OUTSIDE the p-loop: only one pair of B-fragment sets
  // (32 VGPRs) live at a time -> no scratch spills.
#pragma unroll
  for (int t = 0; t < 2; ++t) {
    // B fragments of Wz = wl_z[0:32,:], Wh = wl_h[0:32,:] (H0==0 => rows 32..63 unused)
    // B layout (4x16 f32): VGPR0 = rows k0 / k0+2 (per lane half), VGPR1 = k0+1 / k0+3
    const int col = m + 16 * t;
    v2f BZ[8], BH[8];
#pragma unroll
    for (int ks = 0; ks < 8; ++ks) {
      const int kr = 4 * ks + 2 * khalf;
      BZ[ks].x = wl_z[kr * FF + col]; BZ[ks].y = wl_z[(kr + 1) * FF + col];
      BH[ks].x = wl_h[kr * FF + col]; BH[ks].y = wl_h[(kr + 1) * FF + col];
    }
    const float blz = bl_z[col];
    const float blh = bl_h[col];

#pragma unroll
    for (int p = 0; p < PP; ++p) {
      const float sp   = s[p];
      const float prob = att[p];
      v8f cz = {};
      v8f ch = {};
#pragma unroll
      for (int ks = 0; ks < 8; ++ks) {
        v2f aZ, aH;
        aZ.x = fmaf(sp, wzc[ks].x, bzc[ks].x);
        aZ.y = fmaf(sp, wzc[ks].y, bzc[ks].y);
        aH.x = fmaf(sp, whc[ks].x, bhc[ks].x);
        aH.y = fmaf(sp, whc[ks].y, bhc[ks].y);
        // fp32 WMMA: D = A(16x4) * B(4x16) + C
        cz = __builtin_amdgcn_wmma_f32_16x16x4_f32(false, aZ, false, BZ[ks],
                                                   (short)0, cz, false, false);
        ch = __builtin_amdgcn_wmma_f32_16x16x4_f32(false, aH, false, BH[ks],
                                                   (short)0, ch, false, false);
      }
#pragma unroll
      for (int r = 0; r < 8; ++r) {
        const float z  = 1.0f / (1.0f + __expf(-(cz[r] + blz)));
        const float e2 = __expf(2.0f * (ch[r] + blh));       // tanh via exp
        const float ht = (e2 - 1.0f) / (e2 + 1.0f);
        Hacc[t][r] += prob * (1.0f - z) * ht;                // Hp = Z*0 + (1-Z)*Ht
      }
    }
  }

  // out[node] = b_out + sum_j w_out[j] * relu(H_acc[node][j])
  const float wo0 = w_out[m], wo1 = w_out[m + 16];
  float part[8];
#pragma unroll
  for (int r = 0; r < 8; ++r)
    part[r] = fmaxf(Hacc[0][r], 0.0f) * wo0 + fmaxf(Hacc[1][r], 0.0f) * wo1;
  // reduce the 16 columns held by each 16-lane half (wave32 shuffles)
#pragma unroll
  for (int mask = 1; mask <= 8; mask <<= 1) {
#pragma unroll
    for (int r = 0; r < 8; ++r)
      part[r] += __shfl_xor(part[r], mask, 32);
  }
  if (m == 0) {                          // lane 0 -> rows 0..7, lane 16 -> rows 8..15
    const float bo = b_out[0];
#pragma unroll
    for (int r = 0; r < 8; ++r) {
      const int row = tile * 16 + r + 8 * khalf;
      if (row < n) out[row] = part[r] + bo;
    }
  }
}

extern "C" void kernel_launch(void* const* d_in, const int* in_sizes, int n_in,
                              void* d_out, int out_size, void* d_ws, size_t ws_size,
                              hipStream_t stream) {
  const float* x      = (const float*)d_in[0];
  const int*   ei     = (const int*)d_in[1];
  const float* ew     = (const float*)d_in[2];
  const float* att    = (const float*)d_in[3];
  const float* wc_z   = (const float*)d_in[4];
  const float* bc_z   = (const float*)d_in[5];
  const float* wc_h   = (const float*)d_in[8];
  const float* bc_h   = (const float*)d_in[9];
  const float* wl_z   = (const float*)d_in[10];
  const float* bl_z   = (const float*)d_in[11];
  const float* wl_h   = (const float*)d_in[14];
  const float* bl_h   = (const float*)d_in[15];
  const float* w_out  = (const float*)d_in[16];
  const float* b_out  = (const float*)d_in[17];
  float* out = (float*)d_out;

  const int N = in_sizes[0] / PP;
  const int E = in_sizes[2];

  float* deg  = (float*)d_ws;       // N
  float* dinv = deg + N;            // N
  float* agg  = dinv + N;           // N * 12

  int nbN = (N + 255) / 256;
  int nbE = (E + 255) / 256;
  k_zero<<<nbN, 256, 0, stream>>>(deg, N);
  k_deg<<<nbE, 256, 0, stream>>>(ei, ew, deg, E);
  k_dinv_init<<<nbN, 256, 0, stream>>>(deg, x, dinv, agg, N);
  k_scatter<<<nbE, 256, 0, stream>>>(ei, ew, dinv, x, agg, E);

  const int tiles = (N + 15) / 16;
  const int nbT = (tiles + 7) / 8;  // 8 waves (tiles) per 256-thread block
  k_node<<<nbT, 256, 0, stream>>>(agg, att, wc_z, bc_z, wc_h, bc_h,
                                  wl_z, bl_z, wl_h, bl_h, w_out, b_out,
                                  out, N, tiles);
}